// BahdanauAttention_59605556134637
// MI455X (gfx1250) — compile-verified
//
#include <hip/hip_runtime.h>
#include <hip/hip_bf16.h>
#include <math.h>

// Problem dims (fixed by reference)
#define BATCH 4
#define TQ 256
#define TKDIM 256
#define DDIM 1024
#define UNITS 512
#define NEG_INF -1e9f
#define CU 32  // U-chunk staged in LDS for the score kernel

typedef float v2f __attribute__((ext_vector_type(2)));
typedef float v8f __attribute__((ext_vector_type(8)));

#if __has_builtin(__builtin_amdgcn_tanhf)
#define DEV_TANHF(x) __builtin_amdgcn_tanhf(x)
#else
#define DEV_TANHF(x) tanhf(x)
#endif

// ---------------------------------------------------------------------------
// Generic (batched) f32 GEMM using V_WMMA_F32_16X16X4_F32, register-blocked
// and software-pipelined.
// C[bat] = A[bat] (MxK, row-major, lda=K)  x  B[bat] (KxN, row-major, ldb=N)
//
// Each wave owns a 64x32 block of C: 4 M-tiles x 2 N-tiles = 8 accumulators
// (64 acc VGPRs). Per K-step of 4: 4 A b64 loads + 4 B b32 loads feed 8 WMMAs
// -> 1 VMEM per WMMA, 8 independent accumulation chains. Two fragment register
// sets double-buffer the K-steps so loads for step k+8 are in flight while the
// WMMAs for step k execute (partial s_wait_loadcnt instead of full drains).
// 32-bit unsigned offsets keep addressing in GVS mode (SGPR base + VGPR
// offset). A pairs loaded as v2f vectors -> single global_load_b64 each.
// blockDim=(32,4): 4 waves = 4 N-blocks side by side. Requires K % 8 == 0,
// M % 64 == 0, N % 128 == 0 (true for all three GEMMs here).
//
// ISA VGPR layouts (CDNA5 ISA §7.12.2), f32 16x16x4:
//   A 16x4 : lanes 0-15 row M=lane,    VGPR0=K0, VGPR1=K1
//            lanes 16-31 row M=lane-16, VGPR0=K2, VGPR1=K3
//   B 4x16 : lanes 0-15 col N=lane,    VGPR0=K0, VGPR1=K1
//            lanes 16-31 col N=lane-16, VGPR0=K2, VGPR1=K3
//   C 16x16: VGPR r -> row (r + 8*hi), col (lane&15)
// ---------------------------------------------------------------------------
__global__ __launch_bounds__(128)
void gemm_wmma_f32_16x16x4(const float* __restrict__ A,
                           const float* __restrict__ B,
                           float* __restrict__ C,
                           int M, int N, int K,
                           long sAbat, long sBbat, long sCbat) {
  const int bat = blockIdx.z;
  A += bat * sAbat;
  B += bat * sBbat;
  C += bat * sCbat;

  const int lane = threadIdx.x;  // 0..31
  const int hi   = lane >> 4;    // half-wave select
  const int lo   = lane & 15;
  const int nb   = blockIdx.x * blockDim.y + threadIdx.y;  // 32-wide N block
  const int mb   = blockIdx.y;                             // 64-tall M block
  const unsigned m0 = (unsigned)mb << 6;
  const unsigned n0 = (unsigned)nb << 5;
  const unsigned uN = (unsigned)N;
  const unsigned uK = (unsigned)K;

  // Per-lane fragment offsets (elements), 32-bit on purpose.
  unsigned aoff[4], boff[2];
  #pragma unroll
  for (int i = 0; i < 4; ++i)
    aoff[i] = (m0 + (unsigned)lo + 16u * i) * uK + 2u * (unsigned)hi;
  #pragma unroll
  for (int j = 0; j < 2; ++j)
    boff[j] = 2u * (unsigned)hi * uN + n0 + (unsigned)lo + 16u * j;

  v8f acc[4][2] = {};

  auto loadfrag = [&](v2f (&a)[4], v2f (&b)[2], int k) {
    #pragma unroll
    for (int i = 0; i < 4; ++i) {
      // 8-byte aligned contiguous pair -> one global_load_b64
      a[i] = *(const v2f*)(A + (aoff[i] + (unsigned)k));
    }
    #pragma unroll
    for (int j = 0; j < 2; ++j) {
      const unsigned off = boff[j] + (unsigned)k * uN;
      b[j].x = B[off];
      b[j].y = B[off + uN];     // rows k+2*hi, k+2*hi+1 (stride-N b32 pair)
    }
  };
  auto mmall = [&](v2f (&a)[4], v2f (&b)[2]) {
    #pragma unroll
    for (int i = 0; i < 4; ++i)
      #pragma unroll
      for (int j = 0; j < 2; ++j)
        acc[i][j] = __builtin_amdgcn_wmma_f32_16x16x4_f32(
            false, a[i], false, b[j], (short)0, acc[i][j], false, false);
  };

  // Software pipeline: two fragment sets, prefetch distance = 8 (two K-steps).
  v2f a0[4], b0[2], a1[4], b1[2];
  loadfrag(a0, b0, 0);
  loadfrag(a1, b1, 4);
  int k = 0;
  for (; k + 8 < K; k += 8) {
    mmall(a0, b0);              // consume step k
    loadfrag(a0, b0, k + 8);    // prefetch step k+8
    mmall(a1, b1);              // consume step k+4 (k+8 loads in flight)
    loadfrag(a1, b1, k + 12);   // prefetch step k+12
  }
  mmall(a0, b0);                // epilogue: steps K-8, K-4
  mmall(a1, b1);

  #pragma unroll
  for (int i = 0; i < 4; ++i) {
    #pragma unroll
    for (int j = 0; j < 2; ++j) {
      const unsigned cbase =
          (m0 + 16u * i + 8u * (unsigned)hi) * uN + n0 + 16u * j + (unsigned)lo;
      #pragma unroll
      for (int r = 0; r < 8; ++r) C[cbase + (unsigned)r * uN] = acc[i][j][r];
    }
  }
}

// ---------------------------------------------------------------------------
// Fused additive scores + mask + softmax.
// grid = (TQ/16, B), block = 256 threads (8 waves).
// Thread t owns k-column t for 16 q-rows (16 f32 accumulators).
// U is processed in LDS-staged chunks of CU=32:
//   ldsK[u][k] (transposed, +1 pad -> conflict-free writes, stride-1 reads)
//   ldsQ[q][u] (broadcast reads), ldsS[u] = scale chunk.
// Then row softmax via wave32 shuffle reductions; weights written to attn.
// ---------------------------------------------------------------------------
__global__ __launch_bounds__(256)
void additive_scores_softmax(const float* __restrict__ qbuf,   // [B][TQ][U]
                             const float* __restrict__ kbuf,   // [B][TK][U]
                             const unsigned char* __restrict__ mask, // [B][TK]
                             const float* __restrict__ scale,  // [U]
                             float* __restrict__ attn) {       // [B][TQ][TK]
  __shared__ float ldsK[CU][TKDIM + 1];
  __shared__ float ldsQ[16][CU];
  __shared__ float ldsS[CU];
  __shared__ float ldsScore[16][TKDIM + 1];

  const int b  = blockIdx.y;
  const int q0 = blockIdx.x << 4;
  const int t  = threadIdx.x;  // 0..255 == k index

  const float* kb = kbuf + (long)b * TKDIM * UNITS;
  const float* qb = qbuf + ((long)b * TQ + q0) * UNITS;

  float acc[16];
  #pragma unroll
  for (int i = 0; i < 16; ++i) acc[i] = 0.0f;

  for (int u0 = 0; u0 < UNITS; u0 += CU) {
    __syncthreads();  // previous chunk fully consumed before overwrite
    // Stage k-chunk: coalesced global reads (consec. threads -> consec. u),
    // transposed LDS writes (row stride 257 words -> no bank conflicts).
    for (int idx = t; idx < TKDIM * CU; idx += 256) {
      const int kk = idx / CU;
      const int uu = idx % CU;
      ldsK[uu][kk] = kb[(long)kk * UNITS + u0 + uu];
    }
    for (int idx = t; idx < 16 * CU; idx += 256) {
      const int qq = idx / CU;
      const int uu = idx % CU;
      ldsQ[qq][uu] = qb[(long)qq * UNITS + u0 + uu];
    }
    if (t < CU) ldsS[t] = scale[u0 + t];
    __syncthreads();

    for (int uu = 0; uu < CU; ++uu) {
      const float kval = ldsK[uu][t];   // stride-1 across lanes
      const float su   = ldsS[uu];      // broadcast
      #pragma unroll
      for (int ql = 0; ql < 16; ++ql) {
        acc[ql] += su * DEV_TANHF(ldsQ[ql][uu] + kval);  // broadcast q read
      }
    }
  }

  // Key-side mask (reference: masked -> -1e9 before softmax).
  const bool mk = mask[(long)b * TKDIM + t] != 0;
  #pragma unroll
  for (int ql = 0; ql < 16; ++ql) {
    ldsScore[ql][t] = mk ? acc[ql] : NEG_INF;
  }
  __syncthreads();

  // Row softmax: 8 waves x 2 rows each, 8 columns per lane.
  const int wave = t >> 5;
  const int lane = t & 31;
  for (int row = wave; row < 16; row += 8) {
    float v[8];
    float mx = -INFINITY;
    #pragma unroll
    for (int j = 0; j < 8; ++j) {
      v[j] = ldsScore[row][lane + 32 * j];
      mx = fmaxf(mx, v[j]);
    }
    #pragma unroll
    for (int off = 16; off >= 1; off >>= 1)
      mx = fmaxf(mx, __shfl_xor(mx, off, 32));

    float sum = 0.0f;
    float e[8];
    #pragma unroll
    for (int j = 0; j < 8; ++j) {
      e[j] = __expf(v[j] - mx);
      sum += e[j];
    }
    #pragma unroll
    for (int off = 16; off >= 1; off >>= 1)
      sum += __shfl_xor(sum, off, 32);

    const float inv = 1.0f / sum;
    float* arow = attn + ((long)b * TQ + q0 + row) * TKDIM;
    #pragma unroll
    for (int j = 0; j < 8; ++j) arow[lane + 32 * j] = e[j] * inv;
  }
}

// ---------------------------------------------------------------------------
// Launch: proj-q, proj-k (WMMA GEMM) -> fused scores/softmax -> context (WMMA
// batched GEMM). attn weights are written directly into their d_out slot and
// re-read by the context GEMM (stream-ordered, no extra sync needed).
// ---------------------------------------------------------------------------
extern "C" void kernel_launch(void* const* d_in, const int* in_sizes, int n_in,
                              void* d_out, int out_size, void* d_ws, size_t ws_size,
                              hipStream_t stream) {
  const float*         query = (const float*)d_in[0];         // [4,256,1024]
  const float*         value = (const float*)d_in[1];         // [4,256,1024]
  const unsigned char* mask  = (const unsigned char*)d_in[2]; // [4,256] bool
  const float*         Wq    = (const float*)d_in[3];         // [1024,512]
  const float*         Wk    = (const float*)d_in[4];         // [1024,512]
  const float*         scale = (const float*)d_in[5];         // [512]

  float* out     = (float*)d_out;
  float* context = out;                                 // [4,256,1024]
  float* attn    = out + (long)BATCH * TQ * DDIM;       // [4,256,256]

  float* qbuf = (float*)d_ws;                           // [4*256, 512]
  float* kbuf = qbuf + (long)BATCH * TQ * UNITS;        // [4*256, 512]

  const dim3 blk(32, 4);  // 4 waves; each wave: 64x32 C block

  // q = query·Wq, k = value·Wk   (M=1024, N=512, K=1024, batch folded into M)
  {
    dim3 grid(UNITS / 128, (BATCH * TQ) / 64, 1);       // (4, 16, 1)
    gemm_wmma_f32_16x16x4<<<grid, blk, 0, stream>>>(
        query, Wq, qbuf, BATCH * TQ, UNITS, DDIM, 0, 0, 0);
    gemm_wmma_f32_16x16x4<<<grid, blk, 0, stream>>>(
        value, Wk, kbuf, BATCH * TKDIM, UNITS, DDIM, 0, 0, 0);
  }

  // Fused additive scores + mask + softmax -> attn
  {
    dim3 grid(TQ / 16, BATCH);
    additive_scores_softmax<<<grid, dim3(256), 0, stream>>>(
        qbuf, kbuf, mask, scale, attn);
  }

  // context = attn·value   (per-batch M=256, N=1024, K=256)
  {
    dim3 grid(DDIM / 128, TQ / 64, BATCH);              // (8, 4, 4)
    gemm_wmma_f32_16x16x4<<<grid, blk, 0, stream>>>(
        attn, value, context, TQ, DDIM, TKDIM,
        (long)TQ * TKDIM, (long)TKDIM * DDIM, (long)TQ * DDIM);
  }
}